// MultiHeadAttention_30562987278919
// MI455X (gfx1250) — compile-verified
//
#include <hip/hip_runtime.h>
#include <hip/hip_bf16.h>
#include <math.h>

// Problem constants (from reference): B=4, L=2048, D=512, H=8, E=64
#define BATCH      4
#define SEQ        2048
#define DMODEL     512
#define NHEADS     8
#define HEADDIM    64
#define BH         (BATCH * NHEADS)       // 32
#define MROWS      (BATCH * SEQ)          // 8192
#define SAMPLE_K   40                     // 5 * ceil(ln 2048)
#define NTOP       40
#define NEG_INF_F  (-4294967295.0f)       // -2^32 + 1

typedef float v2f __attribute__((ext_vector_type(2)));
typedef float v8f __attribute__((ext_vector_type(8)));

// ---------------------------------------------------------------------------
// Kernel A: Y = X @ W + bias  for the three projections (blockIdx.z selects).
// X: (8192, 512), W: (512, 512), Y: (8192, 512), all fp32 row-major.
// One wave computes a 16x64 strip of Y using V_WMMA_F32_16X16X4_F32,
// K loop in steps of 4; A fragment reused across 4 N-tiles.
// ---------------------------------------------------------------------------
__global__ __launch_bounds__(256) void proj_gemm_wmma(
    const float* __restrict__ X0, const float* __restrict__ X1, const float* __restrict__ X2,
    const float* __restrict__ W0, const float* __restrict__ W1, const float* __restrict__ W2,
    const float* __restrict__ B0, const float* __restrict__ B1, const float* __restrict__ B2,
    float* __restrict__ Y0, float* __restrict__ Y1, float* __restrict__ Y2)
{
    const float* X; const float* W; const float* bias; float* Y;
    if (blockIdx.z == 0)      { X = X0; W = W0; bias = B0; Y = Y0; }
    else if (blockIdx.z == 1) { X = X1; W = W1; bias = B1; Y = Y1; }
    else                      { X = X2; W = W2; bias = B2; Y = Y2; }

    const int K = DMODEL, N = DMODEL;
    const int lane = threadIdx.x & 31;
    const int wave = threadIdx.x >> 5;
    const int task = blockIdx.x * 8 + wave;      // 512 blocks * 8 waves = 4096 tasks
    const int rowTile  = task >> 3;              // 512 row tiles of 16
    const int colGroup = task & 7;               // 8 col groups of 64
    const int row0 = rowTile << 4;
    const int col0 = colGroup << 6;

    const int m    = lane & 15;                  // A row within tile / C column
    const int half = lane >> 4;                  // lane half selects K pair
    const int ncol = lane & 15;                  // B/C column within 16-tile

    v8f c0 = {}, c1 = {}, c2 = {}, c3 = {};

    const float* arow = X + (size_t)(row0 + m) * K + 2 * half;          // + k0
    const float* bptr = W + (size_t)(2 * half) * N + col0 + ncol;       // + k0*N (+16j)

    for (int k0 = 0; k0 < K; k0 += 4) {
        v2f a;
        a.x = arow[k0];
        a.y = arow[k0 + 1];
        const float* bk = bptr + (size_t)k0 * N;
        v2f b0f, b1f, b2f, b3f;
        b0f.x = bk[0];       b0f.y = bk[N];
        b1f.x = bk[16];      b1f.y = bk[N + 16];
        b2f.x = bk[32];      b2f.y = bk[N + 32];
        b3f.x = bk[48];      b3f.y = bk[N + 48];
        c0 = __builtin_amdgcn_wmma_f32_16x16x4_f32(false, a, false, b0f, (short)0, c0, false, false);
        c1 = __builtin_amdgcn_wmma_f32_16x16x4_f32(false, a, false, b1f, (short)0, c1, false, false);
        c2 = __builtin_amdgcn_wmma_f32_16x16x4_f32(false, a, false, b2f, (short)0, c2, false, false);
        c3 = __builtin_amdgcn_wmma_f32_16x16x4_f32(false, a, false, b3f, (short)0, c3, false, false);
    }

    // C/D layout: VGPR r holds rows (r, r+8) split across lane halves.
    #pragma unroll
    for (int r = 0; r < 8; ++r) {
        const int row = row0 + r + 8 * half;
        float* yr = Y + (size_t)row * N + col0;
        yr[ncol]      = c0[r] + bias[col0 + ncol];
        yr[16 + ncol] = c1[r] + bias[col0 + 16 + ncol];
        yr[32 + ncol] = c2[r] + bias[col0 + 32 + ncol];
        yr[48 + ncol] = c3[r] + bias[col0 + 48 + ncol];
    }
}

// ---------------------------------------------------------------------------
// Kernel B: sampled sparsity measure.
// M[bh,l] = max_s(q_l . k_{idx[l,s]}) - sum_s(q_l . k_{idx[l,s]}) / L
// Qh/Kh viewed as (32, 2048, 64) row-major (flat identity with projection).
// One thread per (bh, l).
// ---------------------------------------------------------------------------
__global__ __launch_bounds__(256) void sample_measure(
    const float* __restrict__ Qh, const float* __restrict__ Kh,
    const int* __restrict__ idxSample, float* __restrict__ Mout)
{
    const int tid = blockIdx.x * blockDim.x + threadIdx.x;   // 65536 total
    const int bh = tid >> 11;
    const int l  = tid & (SEQ - 1);
    const float* qrow = Qh + ((size_t)bh * SEQ + l) * HEADDIM;
    float mx = -INFINITY, sm = 0.0f;
    for (int s = 0; s < SAMPLE_K; ++s) {
        const int ks = idxSample[l * SAMPLE_K + s];
        const float* krow = Kh + ((size_t)bh * SEQ + ks) * HEADDIM;
        float d = 0.0f;
        #pragma unroll 8
        for (int e = 0; e < HEADDIM; ++e) d += qrow[e] * krow[e];
        mx = fmaxf(mx, d);
        sm += d;
    }
    Mout[tid] = mx - sm * (1.0f / (float)SEQ);
}

// ---------------------------------------------------------------------------
// Kernel C: top-40 selection per (bh). One block (256 threads) per bh.
// Iterative argmax with masking; ties resolved to the smaller index.
// ---------------------------------------------------------------------------
__global__ __launch_bounds__(256) void topk_select(
    const float* __restrict__ Mvals, int* __restrict__ topIdx)
{
    __shared__ float vals[SEQ];
    __shared__ float rv[256];
    __shared__ int   ri[256];
    const int bh = blockIdx.x;
    const int t  = threadIdx.x;
    for (int i = 0; i < SEQ / 256; ++i)
        vals[t + 256 * i] = Mvals[(size_t)bh * SEQ + t + 256 * i];
    __syncthreads();

    for (int it = 0; it < NTOP; ++it) {
        float bv = -INFINITY; int bi = SEQ;
        for (int i = 0; i < SEQ / 256; ++i) {
            const int idx = t + 256 * i;
            const float v = vals[idx];
            if (v > bv || (v == bv && idx < bi)) { bv = v; bi = idx; }
        }
        rv[t] = bv; ri[t] = bi;
        __syncthreads();
        for (int s = 128; s > 0; s >>= 1) {
            if (t < s) {
                if (rv[t + s] > rv[t] || (rv[t + s] == rv[t] && ri[t + s] < ri[t])) {
                    rv[t] = rv[t + s]; ri[t] = ri[t + s];
                }
            }
            __syncthreads();
        }
        if (t == 0) {
            topIdx[bh * NTOP + it] = ri[0];
            vals[ri[0]] = -INFINITY;
        }
        __syncthreads();
    }
}

// ---------------------------------------------------------------------------
// Kernel D: full attention for one selected query. One block per (bh, n).
// scores = q . K^T / 8, causal-masked at k > qidx, softmax, out = attn @ V.
// ---------------------------------------------------------------------------
__global__ __launch_bounds__(256) void sparse_attention(
    const float* __restrict__ Qh, const float* __restrict__ Kh,
    const float* __restrict__ Vh, const int* __restrict__ topIdx,
    float* __restrict__ outTop)
{
    __shared__ float qs[HEADDIM];
    __shared__ float p[SEQ];
    __shared__ float red[256];
    __shared__ float part[256];

    const int bh = blockIdx.x / NTOP;
    const int n  = blockIdx.x % NTOP;
    const int t  = threadIdx.x;
    const int qidx = topIdx[bh * NTOP + n];

    if (t < HEADDIM) qs[t] = Qh[((size_t)bh * SEQ + qidx) * HEADDIM + t];
    __syncthreads();

    float sc[SEQ / 256];
    float lmax = -INFINITY;
    for (int i = 0; i < SEQ / 256; ++i) {
        const int k = t + 256 * i;
        const float* krow = Kh + ((size_t)bh * SEQ + k) * HEADDIM;
        float s = 0.0f;
        #pragma unroll 8
        for (int e = 0; e < HEADDIM; ++e) s += qs[e] * krow[e];
        s *= 0.125f;                       // 1/sqrt(64)
        if (k > qidx) s = NEG_INF_F;
        sc[i] = s;
        lmax = fmaxf(lmax, s);
    }
    red[t] = lmax;
    __syncthreads();
    for (int s = 128; s > 0; s >>= 1) {
        if (t < s) red[t] = fmaxf(red[t], red[t + s]);
        __syncthreads();
    }
    const float mx = red[0];
    __syncthreads();

    float lsum = 0.0f;
    for (int i = 0; i < SEQ / 256; ++i) {
        const int k = t + 256 * i;
        const float pe = expf(sc[i] - mx);
        p[k] = pe;
        lsum += pe;
    }
    red[t] = lsum;
    __syncthreads();
    for (int s = 128; s > 0; s >>= 1) {
        if (t < s) red[t] += red[t + s];
        __syncthreads();
    }
    const float inv = 1.0f / red[0];
    __syncthreads();

    // out[e] = sum_k p[k] * V[k][e]; 4 k-quarters x 64 e-lanes.
    const int e  = t & 63;
    const int qq = t >> 6;
    const float* vb = Vh + (size_t)bh * SEQ * HEADDIM;
    float acc = 0.0f;
    for (int k = qq * (SEQ / 4); k < (qq + 1) * (SEQ / 4); ++k)
        acc += p[k] * vb[(size_t)k * HEADDIM + e];
    part[t] = acc;
    __syncthreads();
    if (t < HEADDIM) {
        const float o = (part[t] + part[64 + t] + part[128 + t] + part[192 + t]) * inv;
        outTop[((size_t)bh * NTOP + n) * HEADDIM + t] = o;
    }
}

// ---------------------------------------------------------------------------
// Kernel E: context = cumsum(Vh, axis=l). One thread per (bh, e); serial in l,
// coalesced across the 64 e-lanes. Writes the full output buffer.
// ---------------------------------------------------------------------------
__global__ __launch_bounds__(256) void cumsum_v(
    const float* __restrict__ Vh, float* __restrict__ out)
{
    const int tid = blockIdx.x * blockDim.x + threadIdx.x;   // 2048 total
    const int bh = tid >> 6;
    const int e  = tid & 63;
    const size_t base = (size_t)bh * SEQ * HEADDIM + e;
    float acc = 0.0f;
    for (int l = 0; l < SEQ; ++l) {
        acc += Vh[base + (size_t)l * HEADDIM];
        out[base + (size_t)l * HEADDIM] = acc;
    }
}

// ---------------------------------------------------------------------------
// Kernel F: scatter out_top rows over the cumsum at the selected indices.
// ---------------------------------------------------------------------------
__global__ __launch_bounds__(256) void scatter_top(
    const float* __restrict__ outTop, const int* __restrict__ topIdx,
    float* __restrict__ out)
{
    const int tid = blockIdx.x * blockDim.x + threadIdx.x;   // 32*40*64 = 81920
    const int bh = tid / (NTOP * HEADDIM);
    const int rem = tid % (NTOP * HEADDIM);
    const int n = rem / HEADDIM;
    const int e = rem % HEADDIM;
    const int l = topIdx[bh * NTOP + n];
    out[((size_t)bh * SEQ + l) * HEADDIM + e] = outTop[tid];
}

// ---------------------------------------------------------------------------
extern "C" void kernel_launch(void* const* d_in, const int* in_sizes, int n_in,
                              void* d_out, int out_size, void* d_ws, size_t ws_size,
                              hipStream_t stream) {
    (void)in_sizes; (void)n_in; (void)out_size; (void)ws_size;

    const float* q  = (const float*)d_in[0];
    const float* k  = (const float*)d_in[1];
    const float* v  = (const float*)d_in[2];
    // d_in[3] = mask (unused by the reference computation)
    const float* Wq = (const float*)d_in[4];
    const float* bq = (const float*)d_in[5];
    const float* Wk = (const float*)d_in[6];
    const float* bk = (const float*)d_in[7];
    const float* Wv = (const float*)d_in[8];
    const float* bv = (const float*)d_in[9];
    const int* idxSample = (const int*)d_in[10];
    float* out = (float*)d_out;

    // Workspace layout (bytes)
    char* ws = (char*)d_ws;
    const size_t projBytes = (size_t)MROWS * DMODEL * sizeof(float);   // 16.78 MB
    float* QW     = (float*)(ws);
    float* KW     = (float*)(ws + projBytes);
    float* VW     = (float*)(ws + 2 * projBytes);
    float* Mv     = (float*)(ws + 3 * projBytes);                      // 32*2048 f32
    int*   topIdx = (int*)  (ws + 3 * projBytes + (size_t)BH * SEQ * sizeof(float));
    float* outTop = (float*)(ws + 3 * projBytes + (size_t)BH * SEQ * sizeof(float)
                                + (size_t)BH * NTOP * sizeof(int));

    // 1) Projections (WMMA fp32): 512 blocks x 8 waves x (16x64) tiles, z = q/k/v
    proj_gemm_wmma<<<dim3(512, 1, 3), 256, 0, stream>>>(
        q, k, v, Wq, Wk, Wv, bq, bk, bv, QW, KW, VW);

    // 2) Sampled sparsity measure M: one thread per (bh, l)
    sample_measure<<<(BH * SEQ) / 256, 256, 0, stream>>>(QW, KW, idxSample, Mv);

    // 3) Top-40 per (bh)
    topk_select<<<BH, 256, 0, stream>>>(Mv, topIdx);

    // 4) Attention for selected queries: one block per (bh, n)
    sparse_attention<<<BH * NTOP, 256, 0, stream>>>(QW, KW, VW, topIdx, outTop);

    // 5) cumsum(V) -> output
    cumsum_v<<<(BH * HEADDIM) / 256, 256, 0, stream>>>(VW, out);

    // 6) scatter attention rows over cumsum
    scatter_top<<<(BH * NTOP * HEADDIM) / 256, 256, 0, stream>>>(outTop, topIdx, out);
}